// GCNSampling_18141941859028
// MI455X (gfx1250) — compile-verified
//
#include <hip/hip_runtime.h>

typedef float v2f __attribute__((ext_vector_type(2)));
typedef float v8f __attribute__((ext_vector_type(8)));

#define NL0 512000
#define NL1 102400
#define NL2 20480
#define NL3 4096
#define NE0 1024000
#define NE1 204800
#define NE2 40960

// ------------------------------------------------------------------
// CSR construction: count -> exclusive scan -> cursor fill
// ------------------------------------------------------------------
__global__ void count_edges(const int* __restrict__ dst, int* __restrict__ cnt, int E) {
    int e = blockIdx.x * blockDim.x + threadIdx.x;
    if (e < E) atomicAdd(&cnt[dst[e]], 1);
}

// Single-block multi-pass exclusive scan (n up to ~102K; 100 chunk passes).
__global__ void exclusive_scan(const int* __restrict__ in, int* __restrict__ out, int n) {
    __shared__ int lds[1024];
    const int tid = threadIdx.x;
    int running = 0;
    for (int base = 0; base < n; base += 1024) {
        int i = base + tid;
        int x = (i < n) ? in[i] : 0;
        lds[tid] = x;
        __syncthreads();
        for (int off = 1; off < 1024; off <<= 1) {
            int t = (tid >= off) ? lds[tid - off] : 0;
            __syncthreads();
            lds[tid] += t;
            __syncthreads();
        }
        if (i < n) out[i] = running + lds[tid] - x;   // exclusive
        running += lds[1023];
        __syncthreads();
    }
    if (tid == 0) out[n] = running;
}

__global__ void fill_csr(const int* __restrict__ src, const int* __restrict__ dst,
                         int* __restrict__ cursor, int* __restrict__ eidx, int E) {
    int e = blockIdx.x * blockDim.x + threadIdx.x;
    if (e < E) {
        int p = atomicAdd(&cursor[dst[e]], 1);
        eidx[p] = src[e];
    }
}

// ------------------------------------------------------------------
// Gather-side segment mean: one wave per destination node.
// 32 lanes * float4 = 128 floats (512B) per source row, fully coalesced.
// ------------------------------------------------------------------
template <int F>
__global__ void gather_mean(const float* __restrict__ feats, const int* __restrict__ off,
                            const int* __restrict__ eidx, float* __restrict__ out) {
    const int lane = threadIdx.x & 31;
    const int d    = blockIdx.x * 8 + (threadIdx.x >> 5);
    const int beg  = off[d], end = off[d + 1];
    constexpr int C = F / 128;  // float4-chunks per lane
    float s[C * 4];
#pragma unroll
    for (int i = 0; i < C * 4; ++i) s[i] = 0.f;

    for (int j = beg; j < end; ++j) {
        const float* row = feats + (size_t)eidx[j] * F;
        if (j + 1 < end)  // hide dependent-index latency (global_prefetch_b8)
            __builtin_prefetch(feats + (size_t)eidx[j + 1] * F + lane * 4, 0, 0);
#pragma unroll
        for (int c = 0; c < C; ++c) {
            const float4 v = *reinterpret_cast<const float4*>(row + c * 128 + lane * 4);
            s[c * 4 + 0] += v.x; s[c * 4 + 1] += v.y;
            s[c * 4 + 2] += v.z; s[c * 4 + 3] += v.w;
        }
    }
    const float inv = 1.f / fmaxf((float)(end - beg), 1.f);
    float* o = out + (size_t)d * F;
#pragma unroll
    for (int c = 0; c < C; ++c) {
        float4 v;
        v.x = s[c * 4 + 0] * inv; v.y = s[c * 4 + 1] * inv;
        v.z = s[c * 4 + 2] * inv; v.w = s[c * 4 + 3] * inv;
        *reinterpret_cast<float4*>(o + c * 128 + lane * 4) = v;
    }
}

// ------------------------------------------------------------------
// WMMA helpers (V_WMMA_F32_16X16X4_F32, exact fp32 math, wave32)
// A frag: lanes 0-15 -> (M=lane, K=k0+{0,1}); lanes 16-31 -> (M=lane-16, K=k0+{2,3})
// B frag: mirrored with N across lanes.
// D: vgpr r -> M = r + 8*(lane>=16), N = lane%16.
// ------------------------------------------------------------------
__device__ __forceinline__ v8f wmma4_f32(v2f a, v2f b, v8f c) {
    return __builtin_amdgcn_wmma_f32_16x16x4_f32(false, a, false, b, (short)0, c, false, false);
}

// h[M,128] = relu(h @ W[128,128] + bias)    (in-place; rows wave-disjoint)
__global__ void gemm128_bias_relu(float* __restrict__ hio, const float* __restrict__ W,
                                  const float* __restrict__ bias) {
    const int lane = threadIdx.x & 31;
    const int m0   = (blockIdx.x * 8 + (threadIdx.x >> 5)) * 16;
    const int lm   = lane & 15;
    const int lh   = (lane >> 4) << 1;   // 0 or 2
    const float* arow = hio + (size_t)(m0 + lm) * 128;
    const v8f zero = {0.f, 0.f, 0.f, 0.f, 0.f, 0.f, 0.f, 0.f};
    v8f acc[8];
#pragma unroll
    for (int t = 0; t < 8; ++t) acc[t] = zero;

    for (int k0 = 0; k0 < 128; k0 += 4) {
        v2f a; a.x = arow[k0 + lh]; a.y = arow[k0 + lh + 1];
        const float* wr = W + (size_t)(k0 + lh) * 128;
#pragma unroll
        for (int t = 0; t < 8; ++t) {
            const int n = t * 16 + lm;
            v2f b; b.x = wr[n]; b.y = wr[128 + n];
            acc[t] = wmma4_f32(a, b, acc[t]);
        }
    }
#pragma unroll
    for (int t = 0; t < 8; ++t) {
        const int n  = t * 16 + lm;
        const float bv = bias[n];
#pragma unroll
        for (int r = 0; r < 8; ++r) {
            const int m = m0 + r + ((lane >> 4) << 3);
            hio[(size_t)m * 128 + n] = fmaxf(acc[t][r] + bv, 0.f);
        }
    }
}

// hout[M,256] = concat(z, relu(z)),  z = hin[M,128] @ W + bias
__global__ void gemm128_concat(const float* __restrict__ hin, float* __restrict__ hout,
                               const float* __restrict__ W, const float* __restrict__ bias) {
    const int lane = threadIdx.x & 31;
    const int m0   = (blockIdx.x * 8 + (threadIdx.x >> 5)) * 16;
    const int lm   = lane & 15;
    const int lh   = (lane >> 4) << 1;
    const float* arow = hin + (size_t)(m0 + lm) * 128;
    const v8f zero = {0.f, 0.f, 0.f, 0.f, 0.f, 0.f, 0.f, 0.f};
    v8f acc[8];
#pragma unroll
    for (int t = 0; t < 8; ++t) acc[t] = zero;

    for (int k0 = 0; k0 < 128; k0 += 4) {
        v2f a; a.x = arow[k0 + lh]; a.y = arow[k0 + lh + 1];
        const float* wr = W + (size_t)(k0 + lh) * 128;
#pragma unroll
        for (int t = 0; t < 8; ++t) {
            const int n = t * 16 + lm;
            v2f b; b.x = wr[n]; b.y = wr[128 + n];
            acc[t] = wmma4_f32(a, b, acc[t]);
        }
    }
#pragma unroll
    for (int t = 0; t < 8; ++t) {
        const int n  = t * 16 + lm;
        const float bv = bias[n];
#pragma unroll
        for (int r = 0; r < 8; ++r) {
            const int m = m0 + r + ((lane >> 4) << 3);
            const float z = acc[t][r] + bv;
            hout[(size_t)m * 256 + n]       = z;
            hout[(size_t)m * 256 + 128 + n] = fmaxf(z, 0.f);
        }
    }
}

// out[M,47] = hin[M,256] @ W[256,47] + bias   (N guarded with selects, EXEC stays full)
__global__ void gemm256_final(const float* __restrict__ hin, const float* __restrict__ W,
                              const float* __restrict__ bias, float* __restrict__ out) {
    const int lane = threadIdx.x & 31;
    const int m0   = (blockIdx.x * 8 + (threadIdx.x >> 5)) * 16;
    const int lm   = lane & 15;
    const int lh   = (lane >> 4) << 1;
    const float* arow = hin + (size_t)(m0 + lm) * 256;
    const v8f zero = {0.f, 0.f, 0.f, 0.f, 0.f, 0.f, 0.f, 0.f};
    v8f acc[3];
#pragma unroll
    for (int t = 0; t < 3; ++t) acc[t] = zero;

    for (int k0 = 0; k0 < 256; k0 += 4) {
        v2f a; a.x = arow[k0 + lh]; a.y = arow[k0 + lh + 1];
        const float* wr = W + (size_t)(k0 + lh) * 47;
#pragma unroll
        for (int t = 0; t < 3; ++t) {
            const int n = t * 16 + lm;
            v2f b;
            b.x = (n < 47) ? wr[n] : 0.f;
            b.y = (n < 47) ? wr[47 + n] : 0.f;
            acc[t] = wmma4_f32(a, b, acc[t]);
        }
    }
#pragma unroll
    for (int t = 0; t < 3; ++t) {
        const int n = t * 16 + lm;
        if (n < 47) {
            const float bv = bias[n];
#pragma unroll
            for (int r = 0; r < 8; ++r) {
                const int m = m0 + r + ((lane >> 4) << 3);
                out[(size_t)m * 47 + n] = acc[t][r] + bv;
            }
        }
    }
}

// ------------------------------------------------------------------
extern "C" void kernel_launch(void* const* d_in, const int* in_sizes, int n_in,
                              void* d_out, int out_size, void* d_ws, size_t ws_size,
                              hipStream_t stream) {
    (void)in_sizes; (void)n_in; (void)out_size; (void)ws_size;
    const float* feats = (const float*)d_in[0];
    const int* src0 = (const int*)d_in[1];
    const int* dst0 = (const int*)d_in[2];
    const int* src1 = (const int*)d_in[3];
    const int* dst1 = (const int*)d_in[4];
    const int* src2 = (const int*)d_in[5];
    const int* dst2 = (const int*)d_in[6];
    const float* W1 = (const float*)d_in[7];
    const float* b1 = (const float*)d_in[8];
    const float* W2 = (const float*)d_in[9];
    const float* b2 = (const float*)d_in[10];
    const float* W3 = (const float*)d_in[11];
    const float* b3 = (const float*)d_in[12];

    char* p = (char*)d_ws;
    auto carve = [&](size_t bytes) -> char* {
        char* r = p;
        p += (bytes + 255) & ~(size_t)255;
        return r;
    };
    int*   cnt  = (int*)  carve((size_t)NL1 * 4);
    int*   off  = (int*)  carve((size_t)(NL1 + 1) * 4);
    int*   cur  = (int*)  carve((size_t)(NL1 + 1) * 4);
    int*   eidx = (int*)  carve((size_t)NE0 * 4);
    float* acc0 = (float*)carve((size_t)NL1 * 128 * 4);   // mean feats, then h1 in-place
    float* acc1 = (float*)carve((size_t)NL2 * 128 * 4);
    float* h2   = (float*)carve((size_t)NL2 * 256 * 4);
    float* acc2 = (float*)carve((size_t)NL3 * 256 * 4);

    // ---- layer 0: mean(L0->L1); h1 = relu(mean @ W1 + b1) in-place
    hipMemsetAsync(cnt, 0, (size_t)NL1 * 4, stream);
    count_edges<<<NE0 / 256, 256, 0, stream>>>(dst0, cnt, NE0);
    exclusive_scan<<<1, 1024, 0, stream>>>(cnt, off, NL1);
    hipMemcpyAsync(cur, off, (size_t)NL1 * 4, hipMemcpyDeviceToDevice, stream);
    fill_csr<<<NE0 / 256, 256, 0, stream>>>(src0, dst0, cur, eidx, NE0);
    gather_mean<128><<<NL1 / 8, 256, 0, stream>>>(feats, off, eidx, acc0);
    gemm128_bias_relu<<<NL1 / 128, 256, 0, stream>>>(acc0, W1, b1);

    // ---- layer 1: mean(L1->L2); h2 = concat(z, relu(z)), z = mean @ W2 + b2
    hipMemsetAsync(cnt, 0, (size_t)NL2 * 4, stream);
    count_edges<<<NE1 / 256, 256, 0, stream>>>(dst1, cnt, NE1);
    exclusive_scan<<<1, 1024, 0, stream>>>(cnt, off, NL2);
    hipMemcpyAsync(cur, off, (size_t)NL2 * 4, hipMemcpyDeviceToDevice, stream);
    fill_csr<<<NE1 / 256, 256, 0, stream>>>(src1, dst1, cur, eidx, NE1);
    gather_mean<128><<<NL2 / 8, 256, 0, stream>>>(acc0, off, eidx, acc1);
    gemm128_concat<<<NL2 / 128, 256, 0, stream>>>(acc1, h2, W2, b2);

    // ---- layer 2: mean(L2->L3); out = mean @ W3 + b3
    hipMemsetAsync(cnt, 0, (size_t)NL3 * 4, stream);
    count_edges<<<NE2 / 256, 256, 0, stream>>>(dst2, cnt, NE2);
    exclusive_scan<<<1, 1024, 0, stream>>>(cnt, off, NL3);
    hipMemcpyAsync(cur, off, (size_t)NL3 * 4, hipMemcpyDeviceToDevice, stream);
    fill_csr<<<NE2 / 256, 256, 0, stream>>>(src2, dst2, cur, eidx, NE2);
    gather_mean<256><<<NL3 / 8, 256, 0, stream>>>(h2, off, eidx, acc2);
    gemm256_final<<<NL3 / 128, 256, 0, stream>>>(acc2, W3, b3, (float*)d_out);
}